// LovaszLoss_1176821040028
// MI455X (gfx1250) — compile-verified
//
#include <hip/hip_runtime.h>
#include <hip/hip_bf16.h>
#include <stdint.h>

// ---------------------------------------------------------------------------
// Lovasz softmax-style loss (sigmoid variant), MI455X / gfx1250.
//
// Sort-free formulation: bucket errors by monotone float-bit key (exp + 9
// mantissa bits). Within an equal-valued bucket the Lovasz sum telescopes:
//   sum_k e * (jac_k - jac_{k-1}) = e_bucket * (jac_end - jac_start),
// and jac at bucket boundaries depends only on cumulative (count, positives).
// So: histogram (L2-resident, 21 MB << 192 MB L2) + descending scan.
// HBM traffic ~= one read of pred (176 MB) + label (8 MB) -> ~8 us @ 23.3 TB/s.
//
// CDNA5 paths: v_wmma_f32_16x16x4_f32 (prefix scan as triangular matmul,
// wave32, EXEC all-ones), global_prefetch_b8 via __builtin_prefetch.
// ---------------------------------------------------------------------------

typedef float v2f __attribute__((ext_vector_type(2)));
typedef float v8f __attribute__((ext_vector_type(8)));

namespace {
constexpr int    kB      = 8;
constexpr int    kC      = 21;           // channels; classes 1..20 are scored
constexpr int    kHW     = 512 * 512;    // 262144
constexpr long   kNPix   = (long)kB * kHW;   // 2097152
constexpr int    kNCls   = kC - 1;       // 20
constexpr int    kShift  = 14;           // float bits >> 14: exp + 9 mantissa
constexpr int    kBins   = 65536;        // keys < 0x3F800000>>14 = 65024
constexpr int    kChunk  = 32;
constexpr int    kNChunk = kBins / kChunk;   // 2048
constexpr float  kClampLo = 1e-4f;
constexpr float  kClampHi = 1.0f - 1e-4f;
constexpr double kInvScale = 1.0 / 4294967296.0;   // 2^-32 fixed point
constexpr long   kZeroWords = (long)kNCls * kBins * 4 + kNCls * 2; // cnt+pos+esum+contrib
}

// --------------------------- zero workspace --------------------------------
__global__ void lovasz_zero(uint32_t* __restrict__ w, long nwords) {
    long i = (long)blockIdx.x * blockDim.x + threadIdx.x;
    long stride = (long)gridDim.x * blockDim.x;
    for (; i < nwords; i += stride) w[i] = 0u;
}

// ----------------- phase 1: fused sigmoid/error/histogram ------------------
__global__ __launch_bounds__(256) void lovasz_phase1(
        const float* __restrict__ pred, const int* __restrict__ label,
        uint32_t* __restrict__ cnt, uint32_t* __restrict__ pos,
        unsigned long long* __restrict__ esum) {
    long n = (long)blockIdx.x * blockDim.x + threadIdx.x;
    if (n >= kNPix) return;
    int  b  = (int)(n / kHW);
    int  hw = (int)(n % kHW);
    int  lab = label[n];
    const float* pb = pred + (long)b * kC * kHW + hw;
#pragma unroll 4
    for (int c = 1; c < kC; ++c) {
        float x = pb[(long)c * kHW];
        if (c + 1 < kC)   // prefetch next class line (global_prefetch_b8)
            __builtin_prefetch(pb + (long)(c + 1) * kHW, 0, 1);
        float p = 1.0f / (1.0f + __expf(-x));
        p = fminf(fmaxf(p, kClampLo), kClampHi);
        int   m = (lab == c);
        float e = m ? (1.0f - p) : p;          // |mask - p|, p in (0,1)
        unsigned bin = __float_as_uint(e) >> kShift;
        long idx = (long)(c - 1) * kBins + bin;
        atomicAdd(&cnt[idx], 1u);
        if (m) atomicAdd(&pos[idx], 1u);
        // deterministic fixed-point accumulation (e*2^32 exact for f32 e)
        atomicAdd(&esum[idx], (unsigned long long)(e * 4294967296.0f));
    }
}

// ----------- phase 2a: per-chunk totals (chunk = 32 bins, rank order) ------
__global__ __launch_bounds__(256) void lovasz_phase2a(
        const uint32_t* __restrict__ cnt, const uint32_t* __restrict__ pos,
        uint32_t* __restrict__ chunkN, uint32_t* __restrict__ chunkG) {
    int t = blockIdx.x * blockDim.x + threadIdx.x;
    if (t >= kNCls * kNChunk) return;
    int c = t / kNChunk, ch = t % kNChunk;
    // chunk ch covers descending-error ranks [ch*32, ch*32+32):
    // bin = (kBins-1) - rank
    int binHi = kBins - 1 - ch * kChunk;
    unsigned sn = 0, sg = 0;
    for (int i = 0; i < kChunk; ++i) {
        long idx = (long)c * kBins + (binHi - i);
        sn += cnt[idx];
        sg += pos[idx];
    }
    chunkN[t] = sn;
    chunkG[t] = sg;
}

// ---------------- WMMA-based wave32 inclusive prefix scan ------------------
// Inclusive scan of 16 values as a triangular matmul on the matrix core:
//   D[i][j] = sum_m A[i][m]*B[m][j],  A[i][m] = x_m (bcast over rows),
//   B[m][j] = (m <= j),  accumulated over 4 chained K=4 WMMAs.
// Result: D[i][j] = scan_j for all i -> lands in D VGPR0 = scan_{lane&15}.
// Layout per ISA 7.12.2: A 16x4: v0/v1 = K0/K1 (lanes 0-15), K2/K3 (16-31);
// B 4x16 mirrored; D VGPR0 = (M=0,N=lane | M=8,N=lane-16).
__device__ __forceinline__ float lovasz_wmma_scan16(float x, int base, int lane) {
    v8f d = {0.f, 0.f, 0.f, 0.f, 0.f, 0.f, 0.f, 0.f};
    bool hi = lane >= 16;
    int  j  = lane & 15;
#pragma unroll
    for (int s = 0; s < 4; ++s) {
        float x0 = __shfl(x, base + 4 * s + 0);
        float x1 = __shfl(x, base + 4 * s + 1);
        float x2 = __shfl(x, base + 4 * s + 2);
        float x3 = __shfl(x, base + 4 * s + 3);
        v2f a, bm;
        int k0 = 4 * s + (hi ? 2 : 0);
        a.x  = hi ? x2 : x0;
        a.y  = hi ? x3 : x1;
        bm.x = (k0     <= j) ? 1.0f : 0.0f;
        bm.y = (k0 + 1 <= j) ? 1.0f : 0.0f;
        d = __builtin_amdgcn_wmma_f32_16x16x4_f32(
                /*neg_a=*/false, a, /*neg_b=*/false, bm,
                /*c_mod=*/(short)0, d, /*reuse_a=*/false, /*reuse_b=*/false);
    }
    return d[0];   // scan_{lane & 15}
}

__device__ __forceinline__ float lovasz_wmma_scan32(float x, int lane) {
    float rL = lovasz_wmma_scan16(x, 0,  lane);   // scan of lanes 0..15
    float rH = lovasz_wmma_scan16(x, 16, lane);   // scan of lanes 16..31
    float totL = __shfl(rL, 15);
    return (lane < 16) ? rL : (totL + rH);
}

// -------- phase 2b: serial scan of chunk totals (1 wave per class) ---------
__global__ __launch_bounds__(32) void lovasz_phase2b(
        const uint32_t* __restrict__ chunkN, const uint32_t* __restrict__ chunkG,
        uint32_t* __restrict__ carryK, uint32_t* __restrict__ carryP,
        uint32_t* __restrict__ Gtot) {
    int c = blockIdx.x;
    int lane = threadIdx.x;
    float ck = 0.f, cp = 0.f;   // running exclusive carries (exact: ints < 2^24)
    for (int t0 = 0; t0 < kNChunk; t0 += 32) {
        long t = (long)c * kNChunk + t0 + lane;
        float n = (float)chunkN[t];
        float g = (float)chunkG[t];
        float sn = lovasz_wmma_scan32(n, lane);   // wave-inclusive scan (WMMA)
        float sg = lovasz_wmma_scan32(g, lane);
        carryK[t] = (uint32_t)(ck + sn - n);      // exclusive prefix
        carryP[t] = (uint32_t)(cp + sg - g);
        ck += __shfl(sn, 31);
        cp += __shfl(sg, 31);
    }
    if (lane == 0) Gtot[c] = (uint32_t)cp;        // total positives for class
}

// -------- phase 2c: per-chunk Jaccard telescoping with known carries -------
__device__ __forceinline__ float lovasz_jac(float k, float P, float G) {
    // jac(k) = 1 - (G - P_k) / (G + k - P_k);  jac(0) = 0 when G > 0
    return 1.0f - (G - P) / (G + k - P);
}

__global__ __launch_bounds__(256) void lovasz_phase2c(
        const uint32_t* __restrict__ cnt, const uint32_t* __restrict__ pos,
        const unsigned long long* __restrict__ esum,
        const uint32_t* __restrict__ carryK, const uint32_t* __restrict__ carryP,
        const uint32_t* __restrict__ Gtot,
        unsigned long long* __restrict__ contrib) {
    int t = blockIdx.x * blockDim.x + threadIdx.x;
    if (t >= kNCls * kNChunk) return;
    int c = t / kNChunk, ch = t % kNChunk;
    float G = (float)Gtot[c];
    if (G <= 0.f) return;                      // class absent -> no contribution
    float k = (float)carryK[t];
    float P = (float)carryP[t];
    float jacPrev = lovasz_jac(k, P, G);
    int binHi = kBins - 1 - ch * kChunk;
    double acc = 0.0;
    for (int i = 0; i < kChunk; ++i) {
        long idx = (long)c * kBins + (binHi - i);
        unsigned n = cnt[idx];
        if (!n) continue;                      // empty bucket: jac unchanged
        unsigned g = pos[idx];
        unsigned long long es = esum[idx];
        k += (float)n;
        P += (float)g;
        float jac = lovasz_jac(k, P, G);       // non-decreasing in rank
        double emean = (double)es * kInvScale / (double)n;
        acc += emean * (double)(jac - jacPrev);
        jacPrev = jac;
    }
    if (acc > 0.0)                             // deterministic fixed-point sum
        atomicAdd(&contrib[c], (unsigned long long)(acc * 4294967296.0));
}

// ------------------- phase 3: present-weighted mean ------------------------
__global__ __launch_bounds__(32) void lovasz_phase3(
        const unsigned long long* __restrict__ contrib,
        const uint32_t* __restrict__ Gtot, float* __restrict__ out) {
    int lane = threadIdx.x;
    float v = 0.f, p = 0.f;
    if (lane < kNCls) {
        p = (Gtot[lane] > 0u) ? 1.0f : 0.0f;
        v = p * (float)((double)contrib[lane] * kInvScale);
    }
#pragma unroll
    for (int o = 16; o > 0; o >>= 1) {
        v += __shfl_down(v, o);
        p += __shfl_down(p, o);
    }
    if (lane == 0) out[0] = (p > 0.f) ? (v / p) : 0.f;
}

// ---------------------------------------------------------------------------
extern "C" void kernel_launch(void* const* d_in, const int* in_sizes, int n_in,
                              void* d_out, int out_size, void* d_ws, size_t ws_size,
                              hipStream_t stream) {
    const float* pred  = (const float*)d_in[0];   // [8,21,512,512] f32
    const int*   label = (const int*)d_in[1];     // [8,512,512] i32
    float*       out   = (float*)d_out;           // scalar f32

    // Workspace layout (zeroed region first; ~21.6 MB total)
    uint8_t* w = (uint8_t*)d_ws;
    uint32_t* cnt = (uint32_t*)w;                 w += (size_t)kNCls * kBins * 4;
    uint32_t* pos = (uint32_t*)w;                 w += (size_t)kNCls * kBins * 4;
    unsigned long long* esum = (unsigned long long*)w;   w += (size_t)kNCls * kBins * 8;
    unsigned long long* contrib = (unsigned long long*)w; w += (size_t)kNCls * 8;
    uint32_t* chunkN = (uint32_t*)w;              w += (size_t)kNCls * kNChunk * 4;
    uint32_t* chunkG = (uint32_t*)w;              w += (size_t)kNCls * kNChunk * 4;
    uint32_t* carryK = (uint32_t*)w;              w += (size_t)kNCls * kNChunk * 4;
    uint32_t* carryP = (uint32_t*)w;              w += (size_t)kNCls * kNChunk * 4;
    uint32_t* Gtot   = (uint32_t*)w;              w += (size_t)kNCls * 4;

    lovasz_zero<<<2048, 256, 0, stream>>>((uint32_t*)d_ws, kZeroWords);

    lovasz_phase1<<<(int)(kNPix / 256), 256, 0, stream>>>(pred, label, cnt, pos, esum);

    int nChunkThreads = kNCls * kNChunk;                     // 40960
    lovasz_phase2a<<<(nChunkThreads + 255) / 256, 256, 0, stream>>>(cnt, pos, chunkN, chunkG);

    lovasz_phase2b<<<kNCls, 32, 0, stream>>>(chunkN, chunkG, carryK, carryP, Gtot);

    lovasz_phase2c<<<(nChunkThreads + 255) / 256, 256, 0, stream>>>(
        cnt, pos, esum, carryK, carryP, Gtot, contrib);

    lovasz_phase3<<<1, 32, 0, stream>>>(contrib, Gtot, out);
}